// CustomContrastiveLoss_10969346474222
// MI455X (gfx1250) — compile-verified
//
#include <hip/hip_runtime.h>

#define N_TOK 8192
#define DDIM  256
#define INVALID_CODE (-2147483647 - 1)
#define NCOLT (N_TOK / 64)     // 128 column tiles of 64

typedef __attribute__((ext_vector_type(2))) float v2f;
typedef __attribute__((ext_vector_type(8))) float v8f;

// ---- kernel 1: L2-normalize rows of logits->qn and labels->kn, build adv codes ----
__global__ void __launch_bounds__(256)
normalize_kernel(const float* __restrict__ logits, const float* __restrict__ labels,
                 const int* __restrict__ pad, const int* __restrict__ ad,
                 float* __restrict__ qn, float* __restrict__ kn,
                 int* __restrict__ adv, float* __restrict__ accum) {
    int tid  = threadIdx.x;
    int lane = tid & 31;
    int wv   = tid >> 5;
    int row  = blockIdx.x * 8 + wv;   // 8 waves per block, one row per wave

    if (blockIdx.x == 0 && tid == 0) { accum[0] = 0.0f; accum[1] = 0.0f; }

#pragma unroll
    for (int p = 0; p < 2; ++p) {
        const float* src = (p ? labels : logits) + (size_t)row * DDIM + lane * 8;
        float*       dst = (p ? kn     : qn    ) + (size_t)row * DDIM + lane * 8;
        float4 x0 = *(const float4*)(src);
        float4 x1 = *(const float4*)(src + 4);
        float ss = x0.x*x0.x + x0.y*x0.y + x0.z*x0.z + x0.w*x0.w
                 + x1.x*x1.x + x1.y*x1.y + x1.z*x1.z + x1.w*x1.w;
#pragma unroll
        for (int m = 1; m < 32; m <<= 1) ss += __shfl_xor(ss, m, 32);
        float sc = 1.0f / fmaxf(sqrtf(ss), 1e-12f);
        x0.x *= sc; x0.y *= sc; x0.z *= sc; x0.w *= sc;
        x1.x *= sc; x1.y *= sc; x1.z *= sc; x1.w *= sc;
        *(float4*)(dst)     = x0;
        *(float4*)(dst + 4) = x1;
    }
    if (lane == 0) adv[row] = pad[row] ? ad[row] : INVALID_CODE;
}

// ---- kernel 2: fused WMMA sim + streaming masked softmax sums ----
// Block = 4 waves, owns one 16-row tile. Each wave computes 16x64 output tiles
// (4 column sub-tiles sharing one A operand -> 4 independent WMMA chains),
// striding over the 128 column tiles by 4.
// Since |dot| <= 1, exp2(dot * log2e/tau) in [2^-29, 2^29]: sums cannot
// overflow fp32, so no max-shift is needed -> plain masked sum of exp2.
__global__ void __launch_bounds__(128)
loss_kernel(const float* __restrict__ qn, const float* __restrict__ kn,
            const int* __restrict__ adv, float* __restrict__ accum) {
    __shared__ float red[4][16][2];

    int tid  = threadIdx.x;
    int lane = tid & 31;
    int wv   = tid >> 5;
    int h    = lane >> 4;      // half-wave: K sub-pair for A/B, row group for C
    int nl   = lane & 15;      // M index (A rows) / N index (B cols)
    int row0 = blockIdx.x * 16;

    // A operand (16x4 f32): lane nl supplies row (row0+nl); halves carry K {0,1}/{2,3}
    const float* aptr = qn + (size_t)(row0 + nl) * DDIM + 2 * h;

    // C layout: lane holds rows 8*h + v (v = 0..7), one column per sub-tile
    int  rcode[8];
    bool rval[8];
#pragma unroll
    for (int v = 0; v < 8; ++v) {
        rcode[v] = adv[row0 + 8 * h + v];
        rval[v]  = rcode[v] != INVALID_CODE;
    }

    float sP[8], sA[8];
#pragma unroll
    for (int v = 0; v < 8; ++v) { sP[v] = 0.0f; sA[v] = 0.0f; }

    for (int t = wv; t < NCOLT; t += 4) {
        int col0 = t * 64;
        const float* bptr = kn + (size_t)(col0 + nl) * DDIM + 2 * h;

        // prefetch next tile's B (64KB) so it streams in under this tile's WMMAs
        int tn = t + 4;
        if (tn < NCOLT) {
            const float* pf = kn + (size_t)(tn * 64 + nl) * DDIM;
#pragma unroll
            for (int j = 0; j < 4; ++j)
#pragma unroll
                for (int d = 0; d < 8; ++d)
                    __builtin_prefetch(pf + j * 16 * DDIM + d * 32, 0, 3);
        }

        v8f c0 = {0.f,0.f,0.f,0.f,0.f,0.f,0.f,0.f};
        v8f c1 = {0.f,0.f,0.f,0.f,0.f,0.f,0.f,0.f};
        v8f c2 = {0.f,0.f,0.f,0.f,0.f,0.f,0.f,0.f};
        v8f c3 = {0.f,0.f,0.f,0.f,0.f,0.f,0.f,0.f};

#pragma unroll 8
        for (int s = 0; s < 64; ++s) {
            v2f a  = *(const v2f*)(aptr + 4 * s);
            v2f b0 = *(const v2f*)(bptr + 4 * s);
            v2f b1 = *(const v2f*)(bptr + 16 * DDIM + 4 * s);
            v2f b2 = *(const v2f*)(bptr + 32 * DDIM + 4 * s);
            v2f b3 = *(const v2f*)(bptr + 48 * DDIM + 4 * s);
            c0 = __builtin_amdgcn_wmma_f32_16x16x4_f32(false, a, false, b0, (short)0, c0, false, false);
            c1 = __builtin_amdgcn_wmma_f32_16x16x4_f32(false, a, false, b1, (short)0, c1, false, false);
            c2 = __builtin_amdgcn_wmma_f32_16x16x4_f32(false, a, false, b2, (short)0, c2, false, false);
            c3 = __builtin_amdgcn_wmma_f32_16x16x4_f32(false, a, false, b3, (short)0, c3, false, false);
        }

        int cc0 = adv[col0 + nl];
        int cc1 = adv[col0 + 16 + nl];
        int cc2 = adv[col0 + 32 + nl];
        int cc3 = adv[col0 + 48 + nl];
#pragma unroll
        for (int v = 0; v < 8; ++v) {
            // one exp2 per sim element, shared by both masks
            float e0 = exp2f(c0[v] * 28.853900817779268f);
            float e1 = exp2f(c1[v] * 28.853900817779268f);
            float e2 = exp2f(c2[v] * 28.853900817779268f);
            float e3 = exp2f(c3[v] * 28.853900817779268f);
            bool v0 = cc0 != INVALID_CODE, v1 = cc1 != INVALID_CODE;
            bool v2 = cc2 != INVALID_CODE, v3 = cc3 != INVALID_CODE;
            sA[v] += ((v0 && rval[v]) ? e0 : 0.0f) + ((v1 && rval[v]) ? e1 : 0.0f)
                   + ((v2 && rval[v]) ? e2 : 0.0f) + ((v3 && rval[v]) ? e3 : 0.0f);
            sP[v] += ((v0 && cc0 == rcode[v]) ? e0 : 0.0f) + ((v1 && cc1 == rcode[v]) ? e1 : 0.0f)
                   + ((v2 && cc2 == rcode[v]) ? e2 : 0.0f) + ((v3 && cc3 == rcode[v]) ? e3 : 0.0f);
        }
    }

    // sum across the 16 lanes of each half (lane bits 0..3)
#pragma unroll
    for (int v = 0; v < 8; ++v) {
#pragma unroll
        for (int m = 1; m < 16; m <<= 1) {
            sP[v] += __shfl_xor(sP[v], m, 32);
            sA[v] += __shfl_xor(sA[v], m, 32);
        }
    }

    // one lane per half publishes its 8 rows
    if (nl == 0) {
#pragma unroll
        for (int v = 0; v < 8; ++v) {
            int r = 8 * h + v;
            red[wv][r][0] = sP[v];
            red[wv][r][1] = sA[v];
        }
    }
    __syncthreads();

    // threads 0..15: combine the 4 waves' partials for row (row0+tid)
    if (tid < 16) {
        float SP = 0.0f, SA = 0.0f;
#pragma unroll
        for (int w = 0; w < 4; ++w) { SP += red[w][tid][0]; SA += red[w][tid][1]; }
        bool  has  = SP > 0.0f;   // diagonal guarantees a positive for valid rows
        float loss = has ? (log2f(SA) - log2f(SP)) * 0.69314718055994531f : 0.0f;
        float cnt  = has ? 1.0f : 0.0f;
#pragma unroll
        for (int m = 1; m < 16; m <<= 1) {
            loss += __shfl_xor(loss, m, 32);
            cnt  += __shfl_xor(cnt,  m, 32);
        }
        if (tid == 0) { atomicAdd(accum, loss); atomicAdd(accum + 1, cnt); }
    }
}

__global__ void finalize_kernel(const float* __restrict__ accum, float* __restrict__ out) {
    out[0] = accum[0] / fmaxf(accum[1], 1.0f);
}

extern "C" void kernel_launch(void* const* d_in, const int* in_sizes, int n_in,
                              void* d_out, int out_size, void* d_ws, size_t ws_size,
                              hipStream_t stream) {
    const float* logits = (const float*)d_in[0];
    const float* labels = (const float*)d_in[1];
    const int*   pad    = (const int*)d_in[2];
    const int*   ad     = (const int*)d_in[3];
    float*       out    = (float*)d_out;

    char* ws = (char*)d_ws;
    float* qn    = (float*)ws;                                          // 8 MB
    float* kn    = (float*)(ws + (size_t)N_TOK * DDIM * 4);             // 8 MB
    int*   adv   = (int*)  (ws + (size_t)2 * N_TOK * DDIM * 4);         // 32 KB
    float* accum = (float*)(ws + (size_t)2 * N_TOK * DDIM * 4 + (size_t)N_TOK * 4);

    normalize_kernel<<<N_TOK / 8, 256, 0, stream>>>(logits, labels, pad, ad, qn, kn, adv, accum);
    loss_kernel<<<N_TOK / 16, 128, 0, stream>>>(qn, kn, adv, accum);
    finalize_kernel<<<1, 1, 0, stream>>>(accum, out);
}